// GAT_84043920048227
// MI455X (gfx1250) — compile-verified
//
#include <hip/hip_runtime.h>

#define N_NODES 50000
#define N_EDGES 800000
#define EA (N_EDGES + N_NODES)     // edges + self loops = 850000
#define NUM_GRAPHS 256
#define OUT_C 64
#define NEG_SLOPE 0.2f

typedef __attribute__((ext_vector_type(2))) float v2f;
typedef __attribute__((ext_vector_type(8))) float v8f;

// ---------------- fp32 WMMA GEMM: out[N, NT*16] = A[N, K] * W[K, NT*16] ----
// One wave computes a full 16-row x (NT*16)-col strip: the A fragment is
// loaded once per K-step (single global_load_b64) and reused across NT
// V_WMMA_F32_16X16X4_F32 issues (8 wmma per 17 loads for NT=8).
// A layout (16x4 f32): lanes 0-15 rows, VGPR0/1 = K {kk,kk+1}; lanes 16-31 = K {kk+2,kk+3}.
// B layout (4x16 f32): row K striped across lanes within a VGPR (mirror of A).
// D layout: VGPR j -> row j (lanes 0-15) and row j+8 (lanes 16-31).
template <int NT>
__global__ __launch_bounds__(32) void wmma_gemm_f32(
    const float* __restrict__ A, const float* __restrict__ W,
    float* __restrict__ out, int K) {
  constexpr int Ncols = NT * 16;
  const int m0   = blockIdx.x * 16;
  const int lane = threadIdx.x & 31;
  const int half = lane >> 4;       // 0: K pair {0,1}, 1: K pair {2,3}
  const int lr   = lane & 15;
  const float* __restrict__ arow = A + (size_t)(m0 + lr) * K + 2 * half;
  const float* __restrict__ wcol = W + lr;
  v8f c[NT];
#pragma unroll
  for (int t = 0; t < NT; ++t) c[t] = (v8f){};
  for (int kk = 0; kk < K; kk += 4) {
    const v2f a = *(const v2f*)(arow + kk);              // one b64 load
    const float* __restrict__ w0 = wcol + (size_t)(kk + 2 * half) * Ncols;
    const float* __restrict__ w1 = w0 + Ncols;
#pragma unroll
    for (int t = 0; t < NT; ++t) {
      v2f b;
      b.x = w0[16 * t];
      b.y = w1[16 * t];
      c[t] = __builtin_amdgcn_wmma_f32_16x16x4_f32(
          /*neg_a=*/false, a, /*neg_b=*/false, b,
          /*c_mod=*/(short)0, c[t], /*reuse_a=*/false, /*reuse_b=*/false);
    }
  }
#pragma unroll
  for (int t = 0; t < NT; ++t) {
#pragma unroll
    for (int j = 0; j < 8; ++j) {
      out[(size_t)(m0 + j + 8 * half) * Ncols + 16 * t + lr] = c[t][j];
    }
  }
}

// ---------------- per-(node,head) attention dots --------------------------
__global__ void att_dots(const float* __restrict__ h,
                         const float* __restrict__ att_s,
                         const float* __restrict__ att_d,
                         float* __restrict__ asrc, float* __restrict__ adst,
                         int H, int C) {
  const int i = blockIdx.x * blockDim.x + threadIdx.x;   // node*H + head
  if (i >= N_NODES * H) return;
  const int hd = i % H;
  const float* __restrict__ hp = h + (size_t)i * C;      // (node*H + hd)*C
  const float* __restrict__ as = att_s + hd * C;
  const float* __restrict__ ad = att_d + hd * C;
  float ds = 0.f, dd = 0.f;
  for (int cc = 0; cc < C; ++cc) { const float v = hp[cc]; ds += v * as[cc]; dd += v * ad[cc]; }
  asrc[i] = ds; adst[i] = dd;
}

// ---------------- helpers -------------------------------------------------
__device__ __forceinline__ void edge_sd(const int* __restrict__ ei, int e, int& s, int& d) {
  if (e < N_EDGES) { s = ei[e]; d = ei[N_EDGES + e]; }
  else             { s = d = e - N_EDGES; }              // self loop
}
__device__ __forceinline__ unsigned fenc(float f) {      // order-preserving f32->u32
  unsigned u = __float_as_uint(f);
  return (u & 0x80000000u) ? ~u : (u | 0x80000000u);
}
__device__ __forceinline__ float fdec(unsigned u) {
  return __uint_as_float((u & 0x80000000u) ? (u & 0x7FFFFFFFu) : ~u);
}

__global__ void zero_f(float* __restrict__ p, int n) {
  const int i = blockIdx.x * blockDim.x + threadIdx.x;
  if (i < n) p[i] = 0.f;
}
__global__ void init_ms(unsigned* __restrict__ m, float* __restrict__ s, int n) {
  const int i = blockIdx.x * blockDim.x + threadIdx.x;
  if (i < n) { m[i] = 0u; s[i] = 0.f; }                  // 0 < enc(any finite)
}

// ---------------- pass 1: logits + segment max ----------------------------
__global__ void edge_logits_max(const int* __restrict__ ei,
                                const float* __restrict__ asrc,
                                const float* __restrict__ adst,
                                float* __restrict__ ebuf,
                                unsigned* __restrict__ m, int H) {
  const int i = blockIdx.x * blockDim.x + threadIdx.x;   // e*H + head
  if (i >= EA * H) return;
  const int e = i / H, hd = i % H;
  int s, d; edge_sd(ei, e, s, d);
  float v = asrc[s * H + hd] + adst[d * H + hd];
  v = v > 0.f ? v : NEG_SLOPE * v;
  ebuf[i] = v;
  atomicMax(&m[d * H + hd], fenc(v));
}

// ---------------- pass 2: exp + segment sum -------------------------------
__global__ void edge_exp_sum(const int* __restrict__ ei,
                             float* __restrict__ ebuf,
                             const unsigned* __restrict__ m,
                             float* __restrict__ ssum, int H) {
  const int i = blockIdx.x * blockDim.x + threadIdx.x;
  if (i >= EA * H) return;
  const int e = i / H, hd = i % H;
  int s, d; edge_sd(ei, e, s, d);
  const float ex = __expf(ebuf[i] - fdec(m[d * H + hd]));
  ebuf[i] = ex;
  atomicAdd(&ssum[d * H + hd], ex);
}

// ---------------- pass 3: alpha-weighted scatter of h[src] ----------------
__global__ void edge_scatter(const int* __restrict__ ei,
                             const float* __restrict__ ebuf,
                             const float* __restrict__ ssum,
                             const float* __restrict__ h,
                             float* __restrict__ agg, int H, int HC) {
  const int groups = HC >> 2;                            // 4 channels / thread
  const int i = blockIdx.x * blockDim.x + threadIdx.x;
  if (i >= EA * groups) return;
  const int e = i / groups, g = i % groups;
  const int c0 = g * 4, C = HC / H, hd = c0 / C;
  int s, d; edge_sd(ei, e, s, d);
  const float alpha = ebuf[e * H + hd] / ssum[d * H + hd];
  const float4 hv = *(const float4*)(h + (size_t)s * HC + c0);
  float* op = agg + (size_t)d * HC + c0;
  atomicAdd(op + 0, hv.x * alpha);
  atomicAdd(op + 1, hv.y * alpha);
  atomicAdd(op + 2, hv.z * alpha);
  atomicAdd(op + 3, hv.w * alpha);
}

// ---------------- pass 4: bias (+ELU) -------------------------------------
__global__ void bias_act(float* __restrict__ agg, const float* __restrict__ bias,
                         int HC, int do_elu) {
  const int i = blockIdx.x * blockDim.x + threadIdx.x;
  if (i >= N_NODES * HC) return;
  float v = agg[i] + bias[i % HC];
  if (do_elu) v = v > 0.f ? v : expm1f(v);
  agg[i] = v;
}

// ---------------- global mean pool ----------------------------------------
__global__ void pool_sum(const int* __restrict__ batch, const float* __restrict__ hin,
                         float* __restrict__ out, float* __restrict__ cnt) {
  const int i = blockIdx.x * blockDim.x + threadIdx.x;
  if (i >= N_NODES * OUT_C) return;
  const int n = i / OUT_C, c = i % OUT_C;
  const int b = batch[n];
  atomicAdd(&out[b * OUT_C + c], hin[i]);
  if (c == 0) atomicAdd(&cnt[b], 1.0f);
}
__global__ void pool_div(float* __restrict__ out, const float* __restrict__ cnt) {
  const int i = blockIdx.x * blockDim.x + threadIdx.x;
  if (i >= NUM_GRAPHS * OUT_C) return;
  out[i] /= fmaxf(cnt[i / OUT_C], 1.0f);
}

// ---------------- orchestration -------------------------------------------
extern "C" void kernel_launch(void* const* d_in, const int* in_sizes, int n_in,
                              void* d_out, int out_size, void* d_ws, size_t ws_size,
                              hipStream_t stream) {
  const float* x     = (const float*)d_in[0];
  const int*   ei    = (const int*)  d_in[1];   // [2, E] row-major
  const int*   batch = (const int*)  d_in[2];
  const float* W0  = (const float*)d_in[3];
  const float* as0 = (const float*)d_in[4];
  const float* ad0 = (const float*)d_in[5];
  const float* b0  = (const float*)d_in[6];
  const float* W1  = (const float*)d_in[7];
  const float* as1 = (const float*)d_in[8];
  const float* ad1 = (const float*)d_in[9];
  const float* b1  = (const float*)d_in[10];
  const float* Wl  = (const float*)d_in[11];
  const float* asl = (const float*)d_in[12];
  const float* adl = (const float*)d_in[13];
  const float* bl  = (const float*)d_in[14];

  float* ws = (float*)d_ws;
  float*    h    = ws;                                 // N*128
  float*    g    = h    + (size_t)N_NODES * 128;       // N*128 (agg / next input)
  float*    asr  = g    + (size_t)N_NODES * 128;       // N*4
  float*    adt  = asr  + (size_t)N_NODES * 4;         // N*4
  float*    ebuf = adt  + (size_t)N_NODES * 4;         // EA*4
  unsigned* mbuf = (unsigned*)(ebuf + (size_t)EA * 4); // N*4
  float*    sbuf = (float*)mbuf + (size_t)N_NODES * 4; // N*4
  float*    cnt  = sbuf + (size_t)N_NODES * 4;         // NUM_GRAPHS

  auto layer = [&](const float* in, const float* W, const float* as_,
                   const float* ad_, const float* bias, int H, int C, int do_elu) {
    const int HC = H * C;
    if (HC == 128)
      wmma_gemm_f32<8><<<N_NODES / 16, 32, 0, stream>>>(in, W, h, 128);
    else
      wmma_gemm_f32<4><<<N_NODES / 16, 32, 0, stream>>>(in, W, h, 128);
    att_dots<<<(N_NODES * H + 255) / 256, 256, 0, stream>>>(h, as_, ad_, asr, adt, H, C);
    init_ms<<<(N_NODES * H + 255) / 256, 256, 0, stream>>>(mbuf, sbuf, N_NODES * H);
    zero_f<<<(N_NODES * HC + 255) / 256, 256, 0, stream>>>(g, N_NODES * HC);
    edge_logits_max<<<(EA * H + 255) / 256, 256, 0, stream>>>(ei, asr, adt, ebuf, mbuf, H);
    edge_exp_sum<<<(EA * H + 255) / 256, 256, 0, stream>>>(ei, ebuf, mbuf, sbuf, H);
    edge_scatter<<<(EA * (HC / 4) + 255) / 256, 256, 0, stream>>>(ei, ebuf, sbuf, h, g, H, HC);
    bias_act<<<(N_NODES * HC + 255) / 256, 256, 0, stream>>>(g, bias, HC, do_elu);
  };

  layer(x, W0, as0, ad0, b0, /*H=*/4, /*C=*/32, /*elu=*/1);   // 128 -> 4x32
  layer(g, W1, as1, ad1, b1, /*H=*/4, /*C=*/32, /*elu=*/1);   // 128 -> 4x32
  layer(g, Wl, asl, adl, bl, /*H=*/1, /*C=*/64, /*elu=*/0);   // 128 -> 64

  float* out = (float*)d_out;
  zero_f<<<(NUM_GRAPHS * OUT_C + 255) / 256, 256, 0, stream>>>(out, NUM_GRAPHS * OUT_C);
  zero_f<<<1, 256, 0, stream>>>(cnt, NUM_GRAPHS);
  pool_sum<<<(N_NODES * OUT_C + 255) / 256, 256, 0, stream>>>(batch, g, out, cnt);
  pool_div<<<(NUM_GRAPHS * OUT_C + 255) / 256, 256, 0, stream>>>(out, cnt);
}